// Plane_t_46119358824509
// MI455X (gfx1250) — compile-verified
//
#include <hip/hip_runtime.h>
#include <stdint.h>

#define NCH    32
#define H_T    150
#define BLOCK  256
#define WAVES_PER_BLOCK 8
#define PTS_PER_WAVE 8
#define PTS_PER_BLOCK (WAVES_PER_BLOCK * PTS_PER_WAVE)   // 64 points per block

struct Planes { const float* p[9]; };

__device__ __forceinline__ void coord2idx(float u, int D, int& i0, int& i1, float& w)
{
    // F.grid_sample(align_corners=True, padding_mode='border')
    float f = (u + 1.0f) * 0.5f * (float)(D - 1);
    f = fminf(fmaxf(f, 0.0f), (float)(D - 1));
    float fl = floorf(f);
    i0 = (int)fl;
    i1 = min(i0 + 1, D - 1);
    w  = f - fl;
}

template<bool HWC>
__device__ __forceinline__ float fetch(const float* __restrict__ p, int W, int c, int y, int x)
{
    if (HWC) return p[(((y * W) + x) << 5) + c];          // (H, W, C) channel-contiguous
    return p[c * (H_T * W) + y * W + x];                  // original (C, H, W)
}

// ---- layout transpose (C,H,W) -> (H,W,C): makes channel gathers coalesced ----
// Source is read once -> non-temporal loads (TH=NT keeps L2 clean for grids).
// Destination is re-read ~4.8 GB worth during sampling -> default RT stores
// so the transposed planes land (and stay) in the 192 MB L2.
__global__ __launch_bounds__(BLOCK)
void transpose_chw_hwc(const float* __restrict__ in, float* __restrict__ outp, int HW)
{
    const int lane = threadIdx.x & 31;
    const int wave = (int)((blockIdx.x * blockDim.x + threadIdx.x) >> 5);
    const int nw   = (int)((gridDim.x * blockDim.x) >> 5);
    const int per  = (HW + nw - 1) / nw;
    int s = wave * per;
    int e = min(s + per, HW);
    for (int pix = s; pix < e; ++pix) {
        float v = __builtin_nontemporal_load(&in[(size_t)lane * HW + pix]);
        outp[((size_t)pix << 5) + lane] = v;
    }
}

// ---- main sampling kernel: wave = point, lane = channel ----
template<bool HWC>
__global__ __launch_bounds__(BLOCK)
void kplanes_sample(const float* __restrict__ pts, Planes pl,
                    float* __restrict__ out, int n)
{
    __shared__ float spts[PTS_PER_BLOCK * 4];
    const int tid  = threadIdx.x;
    const int base = blockIdx.x * PTS_PER_BLOCK;

    // Stage this block's 64 point coordinates into LDS via the CDNA5 async
    // LDS-DMA path (tracked by ASYNCcnt), 16B per point.
    if (tid < PTS_PER_BLOCK && (base + tid) < n) {
        unsigned lds = (unsigned)(uintptr_t)(&spts[tid * 4]);
        unsigned long long ga =
            (unsigned long long)(uintptr_t)(pts + (size_t)(base + tid) * 4);
        asm volatile("global_load_async_to_lds_b128 %0, %1, off"
                     :: "v"(lds), "v"(ga) : "memory");
    }
#if __has_builtin(__builtin_amdgcn_s_wait_asynccnt)
    __builtin_amdgcn_s_wait_asynccnt(0);
#else
    asm volatile("s_wait_asynccnt 0" ::: "memory");
#endif
    __syncthreads();

    const int wave = tid >> 5;
    const int lane = tid & 31;

#pragma unroll 2
    for (int j = 0; j < PTS_PER_WAVE; ++j) {
        const int lp = wave * PTS_PER_WAVE + j;
        const int pt = base + lp;
        if (pt >= n) break;                       // wave-uniform
        const float4 p = *(const float4*)(&spts[lp * 4]);

        // time axis (H=150) shared by all 9 planes
        int y0, y1; float wy;
        coord2idx(p.w, H_T, y0, y1, wy);
        const float cs3[3] = {p.x, p.y, p.z};
        float* o = out + (size_t)pt * (3 * NCH);

#pragma unroll
        for (int si = 0; si < 3; ++si) {
            const int W = 128 << si;
            float acc = 1.0f;
#pragma unroll
            for (int ci = 0; ci < 3; ++ci) {
                int x0, x1; float wx;
                coord2idx(cs3[ci], W, x0, x1, wx);
                const float* g = pl.p[si * 3 + ci];
                float v00 = fetch<HWC>(g, W, lane, y0, x0);
                float v01 = fetch<HWC>(g, W, lane, y0, x1);
                float v10 = fetch<HWC>(g, W, lane, y1, x0);
                float v11 = fetch<HWC>(g, W, lane, y1, x1);
                float top = v00 * (1.0f - wx) + v01 * wx;
                float bot = v10 * (1.0f - wx) + v11 * wx;
                acc *= top * (1.0f - wy) + bot * wy;
            }
            // 384 MB output stream: non-temporal store keeps the L2-resident
            // transposed grids from being evicted by the write stream.
            __builtin_nontemporal_store(acc, &o[si * NCH + lane]);
        }
    }
}

extern "C" void kernel_launch(void* const* d_in, const int* in_sizes, int n_in,
                              void* d_out, int out_size, void* d_ws, size_t ws_size,
                              hipStream_t stream)
{
    const float* pts = (const float*)d_in[0];
    float* out = (float*)d_out;
    const int n = in_sizes[0] / 4;

    Planes pl;
    size_t planeElems[9];
    size_t tot = 0;
    for (int si = 0; si < 3; ++si)
        for (int ci = 0; ci < 3; ++ci) {
            const int k = si * 3 + ci;
            pl.p[k] = (const float*)d_in[1 + k];
            planeElems[k] = (size_t)NCH * H_T * (size_t)(128 << si);
            tot += planeElems[k];
        }

    const int blocks = (n + PTS_PER_BLOCK - 1) / PTS_PER_BLOCK;
    const bool useT = (ws_size >= tot * sizeof(float));

    if (useT) {
        // per-call layout transform into workspace (idempotent, fully
        // overwrites its region -> deterministic across graph replays)
        Planes plT;
        float* ws = (float*)d_ws;
        size_t off = 0;
        for (int k = 0; k < 9; ++k) {
            const int si = k / 3;
            const int HW = H_T * (128 << si);
            transpose_chw_hwc<<<256, BLOCK, 0, stream>>>(pl.p[k], ws + off, HW);
            plT.p[k] = ws + off;
            off += planeElems[k];
        }
        kplanes_sample<true><<<blocks, BLOCK, 0, stream>>>(pts, plT, out, n);
    } else {
        kplanes_sample<false><<<blocks, BLOCK, 0, stream>>>(pts, pl, out, n);
    }
}